// VoronoiSimpleIntegrandSlang_5222680232015
// MI455X (gfx1250) — compile-verified
//
#include <hip/hip_runtime.h>

typedef __attribute__((ext_vector_type(2))) float v2f;
typedef __attribute__((ext_vector_type(8))) float v8f;

#define NSITE_TILES 7   // ceil(100 / 16) site tiles

// One wave handles 16 queries per loop iteration.
//   score(site m, query n) = -2 cx*qx - 2 cy*qy + (cx^2+cy^2)   (= d2 - |q|^2, argmin-equivalent)
// One V_WMMA_F32_16X16X4_F32 per 16-site tile; |c|^2 bias folded in via the K=2 row
// (B supplies 1.0 there), so the matrix op computes the complete affine score.
__global__ __launch_bounds__(256) void voronoi_wmma_kernel(
    const float* __restrict__ x, const float* __restrict__ p,
    float* __restrict__ out, int nQueries, int nSites)
{
    const int  lane  = threadIdx.x & 31;
    const int  sLane = lane & 15;
    const bool hiH   = lane >= 16;

    // ---- Per-tile A matrices (site constants), kept in registers ----
    // A 16x4 f32 layout: lanes 0-15 -> VGPR0=K0, VGPR1=K1 ; lanes 16-31 -> VGPR0=K2, VGPR1=K3
    float a0[NSITE_TILES], a1[NSITE_TILES];
#pragma unroll
    for (int t = 0; t < NSITE_TILES; ++t) {
        int s = t * 16 + sLane;
        float px = 0.0f, py = 0.0f, c2 = 3.0e38f;   // padding sites -> huge score
        if (s < nSites) {
            px = p[5 * s + 0];
            py = p[5 * s + 1];
            c2 = px * px + py * py;
        }
        a0[t] = hiH ? c2   : (-2.0f * px);   // K=0 row: -2cx   | K=2 row: |c|^2
        a1[t] = hiH ? 0.0f : (-2.0f * py);   // K=1 row: -2cy   | K=3 row: 0
    }

    const int wavesPerBlock = blockDim.x >> 5;
    const int waveId = blockIdx.x * wavesPerBlock + (int)(threadIdx.x >> 5);
    const int nWaves = gridDim.x * wavesPerBlock;
    const int nTiles = (nQueries + 15) >> 4;

    for (int tile = waveId; tile < nTiles; tile += nWaves) {
        const int qbase = tile << 4;

        // ---- B 4x16 f32: lanes 0-15 -> K0=qx, K1=qy ; lanes 16-31 -> K2=1, K3=0 ----
        float qx = 0.0f, qy = 0.0f;
        if (lane < 16 && (qbase + lane) < nQueries) {
            qx = x[2 * (qbase + lane) + 0];
            qy = x[2 * (qbase + lane) + 1];
        }
        v2f bmat;
        bmat.x = hiH ? 1.0f : qx;
        bmat.y = hiH ? 0.0f : qy;

        // 4 independent (value, index) accumulators -> 4x shorter dependency chain.
        // Indices are tracked WITHOUT the lane-dependent +8 offset so every
        // index source is a uniform compile-time constant (inline/literal cndmask).
        float bV0 = 3.0e38f, bV1 = 3.0e38f, bV2 = 3.0e38f, bV3 = 3.0e38f;
        int   bI0 = 0, bI1 = 1, bI2 = 2, bI3 = 3;

#pragma unroll
        for (int t = 0; t < NSITE_TILES; ++t) {
            v2f amat;
            amat.x = a0[t];
            amat.y = a1[t];
            v8f cacc = {};   // C = 0
            v8f d = __builtin_amdgcn_wmma_f32_16x16x4_f32(
                false, amat, false, bmat, (short)0, cacc, false, false);
            // D layout: VGPR r, lane L -> score of site (16t + r + 8*(L>=16)) for query (L&15)
#pragma unroll
            for (int r = 0; r < 8; r += 4) {
                const int base = t * 16 + r;
                { float v = d[r + 0]; bool m = v < bV0; bV0 = m ? v : bV0; bI0 = m ? (base + 0) : bI0; }
                { float v = d[r + 1]; bool m = v < bV1; bV1 = m ? v : bV1; bI1 = m ? (base + 1) : bI1; }
                { float v = d[r + 2]; bool m = v < bV2; bV2 = m ? v : bV2; bI2 = m ? (base + 2) : bI2; }
                { float v = d[r + 3]; bool m = v < bV3; bV3 = m ? v : bV3; bI3 = m ? (base + 3) : bI3; }
            }
        }

        // ---- Merge the 4 accumulators (lexicographic: value, then lowest index) ----
        if (bV1 < bV0 || (bV1 == bV0 && bI1 < bI0)) { bV0 = bV1; bI0 = bI1; }
        if (bV3 < bV2 || (bV3 == bV2 && bI3 < bI2)) { bV2 = bV3; bI2 = bI3; }
        if (bV2 < bV0 || (bV2 == bV0 && bI2 < bI0)) { bV0 = bV2; bI0 = bI2; }

        // Apply the lane-half site offset once, then merge the two half-wave ranges
        // (lanes L and L^16 evaluated complementary site halves of the same query).
        float bestV = bV0;
        int   bestI = bI0 + (hiH ? 8 : 0);
        float oV = __shfl_xor(bestV, 16);
        int   oI = __shfl_xor(bestI, 16);
        if (oV < bestV || (oV == bestV && oI < bestI)) { bestI = oI; }

        // ---- Gather winning color and store ----
        if (lane < 16) {
            const int q = qbase + lane;
            if (q < nQueries) {
                const int b5 = 5 * bestI;
                out[3 * q + 0] = p[b5 + 2];
                out[3 * q + 1] = p[b5 + 3];
                out[3 * q + 2] = p[b5 + 4];
            }
        }
    }
}

extern "C" void kernel_launch(void* const* d_in, const int* in_sizes, int n_in,
                              void* d_out, int out_size, void* d_ws, size_t ws_size,
                              hipStream_t stream) {
    (void)n_in; (void)out_size; (void)d_ws; (void)ws_size;
    const float* x = (const float*)d_in[0];   // [N,2] queries
    const float* p = (const float*)d_in[1];   // [K*5] sites (x,y,r,g,b)
    float* out = (float*)d_out;               // [N,3] colors

    const int nQueries = in_sizes[0] / 2;
    const int nSites   = in_sizes[1] / 5;

    const int threads = 256;                  // 8 waves per block (wave32)
    const int wavesPerBlock = threads / 32;
    const int nTiles = (nQueries + 15) / 16;
    int blocks = (nTiles + wavesPerBlock - 1) / wavesPerBlock;
    if (blocks > 4096) blocks = 4096;         // ~4 query-tiles per wave via grid-stride
    if (blocks < 1) blocks = 1;

    voronoi_wmma_kernel<<<blocks, threads, 0, stream>>>(x, p, out, nQueries, nSites);
}